// AddShift_mp_blur_module_60035052863996
// MI455X (gfx1250) — compile-verified
//
#include <hip/hip_runtime.h>
#include <stdint.h>

// ---------------- problem constants (mirror the reference) ----------------
#define B_    4
#define G_    2
#define CO_   64
#define NK_   17
#define H_    56
#define W_    56
#define HW_   (H_ * W_)              // 3136
#define CIN_  (G_ * CO_ * NK_)       // 2176
#define NJ    (G_ * NK_)             // 34 (g,k) pairs per oc

#define TILE_H   4
#define NTILE    (H_ / TILE_H)       // 14
#define THREADS  (TILE_H * W_)       // 224 threads = 7 wave32
#define SROWS    6                   // staged rows per plane: h0-1 .. h0+4
#define CHUNKS_PER_ROW 14            // 56 floats = 14 x b128
#define TOTAL_CHUNKS (NJ * SROWS * CHUNKS_PER_ROW)  // 2856

// LDS byte-offset of a generic pointer to a __shared__ object
__device__ __forceinline__ uint32_t ldsoff(const void* p) {
    return (uint32_t)(uintptr_t)(__attribute__((address_space(3))) const void*)p;
}

// CDNA5 async global -> LDS DMA (GVS mode: saddr base + 32-bit vaddr offset)
__device__ __forceinline__ void async_ld_b32(uint32_t lds, uint32_t goff, const void* base) {
    asm volatile("global_load_async_to_lds_b32 %0, %1, %2"
                 :: "v"(lds), "v"(goff), "s"(base) : "memory");
}
__device__ __forceinline__ void async_ld_b128(uint32_t lds, uint32_t goff, const void* base) {
    asm volatile("global_load_async_to_lds_b128 %0, %1, %2"
                 :: "v"(lds), "v"(goff), "s"(base) : "memory");
}
__device__ __forceinline__ void wait_async0() {
    asm volatile("s_wait_asynccnt 0x0" ::: "memory");
}

__global__ __launch_bounds__(THREADS)
void shift_accum_kernel(const float* __restrict__ x,
                        const int*   __restrict__ pad_hv,     // [CO_*NK_, 4]
                        const int*   __restrict__ idx_id,     // [CO_*NK_, 2]
                        float*       __restrict__ out)
{
    __shared__ float s_x[NJ * SROWS * W_];   // 45,696 B staged input rows
    __shared__ int   s_tab[3 * NJ];          // [0..33]=dx, [34..67]=dy, [68..101]=v

    const int tid  = threadIdx.x;
    const int bx   = blockIdx.x;
    const int tile = bx % NTILE;
    const int bc   = bx / NTILE;
    const int oc   = bc % CO_;
    const int b    = bc / CO_;
    const int h0   = tile * TILE_H;

    const float* xb = x + (size_t)b * CIN_ * HW_;   // batch slab (uniform per block)

    // ---- stage shift tables via async b32 DMA ----
    if (tid < 2 * NJ) {                       // dx (t=0) and dy (t=1) from pad_hv
        int t = tid / NJ;
        int e = tid % NJ;
        int g = e / NK_, k = e % NK_;
        uint32_t goff = (uint32_t)((((oc * NK_ + k) * (2 * G_)) + t * G_ + g) * 4);
        async_ld_b32(ldsoff(&s_tab[tid]), goff, pad_hv);
    } else if (tid < 3 * NJ) {                // v from idx_identit
        int e = tid - 2 * NJ;
        int g = e / NK_, k = e % NK_;
        uint32_t goff = (uint32_t)((((oc * NK_ + k) * G_) + g) * 4);
        async_ld_b32(ldsoff(&s_tab[tid]), goff, idx_id);
    }

    // ---- stage 6 rows x 34 planes via async b128 DMA (rows clamped = replicate) ----
    for (int cid = tid; cid < TOTAL_CHUNKS; cid += THREADS) {
        int j   = cid / (SROWS * CHUNKS_PER_ROW);     // 0..33  (g*17+k)
        int rem = cid % (SROWS * CHUNKS_PER_ROW);
        int rr  = rem / CHUNKS_PER_ROW;               // 0..5
        int q   = rem % CHUNKS_PER_ROW;               // 0..13
        int g = j / NK_, k = j % NK_;
        int srow = min(max(h0 - 1 + rr, 0), H_ - 1);
        uint32_t goff  = (uint32_t)(((((g * CO_ + oc) * NK_ + k) * HW_) + srow * W_ + q * 4) * 4);
        uint32_t laddr = ldsoff(&s_x[(j * SROWS + rr) * W_ + q * 4]);
        async_ld_b128(laddr, goff, xb);
    }

    wait_async0();
    __syncthreads();

    // ---- accumulate the three branches ----
    const int w  = tid % W_;
    const int hl = tid / W_;
    const int h  = h0 + hl;

    float a1 = 0.f, a2 = 0.f, a3 = 0.f;

    #pragma unroll
    for (int g = 0; g < G_; ++g) {
        const float* pg = xb + (size_t)(g * CO_ + oc) * NK_ * HW_;
        #pragma unroll
        for (int k = 0; k < NK_; ++k) {
            const int j = g * NK_ + k;
            const float* s_pl = &s_x[j * SROWS * W_];

            // branch 1: horizontal shift, row h  (from LDS, staged row hl+1)
            {
                int dx = s_tab[j];
                int c  = w + dx;
                if ((unsigned)(c + 1) <= (unsigned)(W_ + 1)) {   // -1 <= c <= 56
                    int cc = min(max(c, 0), W_ - 1);
                    a1 += s_pl[(hl + 1) * W_ + cc];
                }
            }
            // branch 2: vertical shift (scattered rows -> coalesced global load)
            {
                int dy = s_tab[NJ + j];
                int r  = h + dy;
                if ((unsigned)(r + 1) <= (unsigned)(H_ + 1)) {   // -1 <= r <= 56
                    int rc = min(max(r, 0), H_ - 1);
                    a2 += pg[(size_t)k * HW_ + rc * W_ + w];
                }
            }
            // branch 3: masked 3x3 tap (from LDS; staged rows cover h-1..h+1)
            {
                int v = s_tab[2 * NJ + j];
                if (v >= 0) {
                    int ddy = v / 3 - 1;
                    int ddx = v % 3 - 1;
                    int cc  = min(max(w + ddx, 0), W_ - 1);
                    a3 += s_pl[(hl + 1 + ddy) * W_ + cc];
                }
            }
        }
    }

    const size_t o = (((size_t)b * CO_ + oc) * H_ + h) * W_ + w;
    const size_t branch_stride = (size_t)B_ * CO_ * HW_;   // 802816
    out[o]                     = a1;
    out[branch_stride + o]     = a2;
    out[2 * branch_stride + o] = a3;
}

extern "C" void kernel_launch(void* const* d_in, const int* in_sizes, int n_in,
                              void* d_out, int out_size, void* d_ws, size_t ws_size,
                              hipStream_t stream) {
    (void)in_sizes; (void)n_in; (void)out_size; (void)d_ws; (void)ws_size;
    const float* x      = (const float*)d_in[0];
    const int*   pad_hv = (const int*)d_in[1];
    const int*   idx_id = (const int*)d_in[2];
    float*       out    = (float*)d_out;

    dim3 grid(B_ * CO_ * NTILE);   // 4*64*14 = 3584 blocks
    dim3 block(THREADS);           // 224 threads (7 wave32)
    shift_accum_kernel<<<grid, block, 0, stream>>>(x, pad_hv, idx_id, out);
}